// LinearAttention_52063593562940
// MI455X (gfx1250) — compile-verified
//
#include <hip/hip_runtime.h>
#include <hip/hip_bf16.h>

#define NB 4
#define LL 8192
#define HH 16
#define DD 64
#define CHUNK 512
#define SUB 32

typedef __attribute__((ext_vector_type(2))) float v2f;
typedef __attribute__((ext_vector_type(8))) float v8f;

__device__ __forceinline__ float elu1(float x) {
    // elu(x) + 1 == x > 0 ? x + 1 : exp(x)
    return x > 0.0f ? x + 1.0f : __expf(x);
}

// ---------------------------------------------------------------------------
// Zero workspace (A matrices + ksum); harness poisons d_ws with 0xAA.
// ---------------------------------------------------------------------------
__global__ void la_zero(float* __restrict__ ws, int count) {
    int i = blockIdx.x * 256 + threadIdx.x;
    if (i < count) ws[i] = 0.0f;
}

// ---------------------------------------------------------------------------
// Phase 1: per (n,h):  A[d,m] += sum_s V[s,d] * Kf[s,m];  ksum[m] += Kf[s,m]
// grid: (NB*HH, LL/CHUNK), block: 128 (4 waves). Wave w owns d-band [16w,16w+16).
// WMMA f32 16x16x4: Afrag[i,k] = V[s_k, dbase+i], Bfrag[k,j] = Kf[s_k, mbase+j].
// ---------------------------------------------------------------------------
__global__ __launch_bounds__(128) void la_phase1(const float* __restrict__ Kp,
                                                 const float* __restrict__ Vp,
                                                 float* __restrict__ Aws,
                                                 float* __restrict__ KSws) {
    __shared__ float ldsK[SUB][DD];

    const int nh    = blockIdx.x;              // 0..63
    const int chunk = blockIdx.y;              // 0..15
    const int n = nh / HH, h = nh % HH;
    const int lane  = threadIdx.x & 31;
    const int wave  = threadIdx.x >> 5;
    const int row   = lane & 15;
    const int half  = lane >> 4;
    const int khalf = half << 1;               // 0 or 2

    const size_t rowStride = (size_t)HH * DD;  // stride between consecutive s
    const size_t base = (size_t)n * LL * rowStride + (size_t)h * DD;

    v8f acc[4] = {};
    float ks0 = 0.0f, ks1 = 0.0f;
    const int s0 = chunk * CHUNK;
    const int dbase = wave * 16;

    for (int c = 0; c < CHUNK; c += SUB) {
        __syncthreads();
        // Stage Kf tile [SUB][DD] (2048 elems / 128 threads), coalesced in d.
        for (int e = threadIdx.x; e < SUB * DD; e += 128) {
            int ss = e >> 6, d = e & 63;
            ldsK[ss][d] = elu1(Kp[base + (size_t)(s0 + c + ss) * rowStride + d]);
        }
        __syncthreads();

        if (wave == 0) {  // ksum partials: lane owns columns lane and lane+32
            for (int ss = 0; ss < SUB; ++ss) {
                ks0 += ldsK[ss][lane];
                ks1 += ldsK[ss][lane + 32];
            }
        }

        for (int ss = 0; ss < SUB; ss += 4) {
            const int sA = s0 + c + ss + khalf;
            v2f a;  // Afrag: V[s_k, dbase+row]; d contiguous across lanes 0-15
            a.x = Vp[base + (size_t)(sA)     * rowStride + dbase + row];
            a.y = Vp[base + (size_t)(sA + 1) * rowStride + dbase + row];
#pragma unroll
            for (int t = 0; t < 4; ++t) {
                v2f b;  // Bfrag: Kf[s_k, t*16+col]
                b.x = ldsK[ss + khalf    ][t * 16 + row];
                b.y = ldsK[ss + khalf + 1][t * 16 + row];
                acc[t] = __builtin_amdgcn_wmma_f32_16x16x4_f32(
                    false, a, false, b, (short)0, acc[t], false, false);
            }
        }
    }

    // Merge partial A band: C layout row = r + 8*half, col = lane&15.
    float* Am = Aws + (size_t)nh * DD * DD;
#pragma unroll
    for (int t = 0; t < 4; ++t) {
#pragma unroll
        for (int r = 0; r < 8; ++r) {
            int d = dbase + r + 8 * half;
            int m = t * 16 + row;
            atomicAdd(&Am[d * DD + m], acc[t][r]);
        }
    }
    if (wave == 0) {
        atomicAdd(&KSws[nh * DD + lane], ks0);
        atomicAdd(&KSws[nh * DD + lane + 32], ks1);
    }
}

// ---------------------------------------------------------------------------
// Phase 2: out[l,m] = z[l] * sum_d Qf[l,d] * A[d,m],
//          z[l] = 1/(sum_d Qf[l,d]*ksum[d] + eps)
// grid: (NB*HH, LL/64), block 128 (4 waves); wave w owns rows [lb+16w, +16).
// ---------------------------------------------------------------------------
__global__ __launch_bounds__(128) void la_phase2(const float* __restrict__ Qp,
                                                 const float* __restrict__ Aws,
                                                 const float* __restrict__ KSws,
                                                 float* __restrict__ Out) {
    __shared__ float ldsQ[4][16][DD];
    __shared__ float zbuf[4][16];

    const int nh = blockIdx.x;                 // 0..63
    const int lb = blockIdx.y * 64;            // L band base
    const int n = nh / HH, h = nh % HH;
    const int lane  = threadIdx.x & 31;
    const int wave  = threadIdx.x >> 5;
    const int row   = lane & 15;
    const int half  = lane >> 4;
    const int khalf = half << 1;

    const size_t rowStride = (size_t)HH * DD;
    const size_t base = (size_t)n * LL * rowStride + (size_t)h * DD;
    const int lw = lb + wave * 16;

    // Stage Qf tile [16][64] for this wave (coalesced in d, feature map once).
    for (int e = lane; e < 16 * DD; e += 32) {
        int r = e >> 6, d = e & 63;
        ldsQ[wave][r][d] = elu1(Qp[base + (size_t)(lw + r) * rowStride + d]);
    }

    // z per row (lanes 0..15, each owns one row).
    if (lane < 16) {
        const float* ks = KSws + nh * DD;
        float zacc = 0.0f;
        for (int d = 0; d < DD; ++d) zacc += ldsQ[wave][lane][d] * ks[d];
        zbuf[wave][lane] = 1.0f / (zacc + 1e-6f);
    }

    const float* Am = Aws + (size_t)nh * DD * DD;  // resident in L2 (1 MB total)
    v8f acc[4] = {};
#pragma unroll 4
    for (int kd = 0; kd < 16; ++kd) {
        v2f a;  // Afrag: Qf[lw+row, 4*kd + k]
        a.x = ldsQ[wave][row][4 * kd + khalf];
        a.y = ldsQ[wave][row][4 * kd + khalf + 1];
#pragma unroll
        for (int t = 0; t < 4; ++t) {
            v2f b;  // Bfrag: A[4*kd + k, t*16 + col]
            b.x = Am[(4 * kd + khalf)     * DD + t * 16 + row];
            b.y = Am[(4 * kd + khalf + 1) * DD + t * 16 + row];
            acc[t] = __builtin_amdgcn_wmma_f32_16x16x4_f32(
                false, a, false, b, (short)0, acc[t], false, false);
        }
    }

    // Scale by z and store (coalesced: m contiguous across lanes 0-15).
#pragma unroll
    for (int t = 0; t < 4; ++t) {
#pragma unroll
        for (int r = 0; r < 8; ++r) {
            int i = r + 8 * half;
            int l = lw + i;
            float z = zbuf[wave][i];
            Out[base + (size_t)l * rowStride + t * 16 + row] = acc[t][r] * z;
        }
    }
}

// ---------------------------------------------------------------------------
extern "C" void kernel_launch(void* const* d_in, const int* in_sizes, int n_in,
                              void* d_out, int out_size, void* d_ws, size_t ws_size,
                              hipStream_t stream) {
    const float* Q = (const float*)d_in[0];
    const float* K = (const float*)d_in[1];
    const float* V = (const float*)d_in[2];
    float* out = (float*)d_out;

    float* Aws = (float*)d_ws;                       // NB*HH*64*64 floats = 1 MB
    float* KS  = Aws + (size_t)NB * HH * DD * DD;    // NB*HH*64 floats

    const int zc = NB * HH * DD * DD + NB * HH * DD;
    la_zero<<<(zc + 255) / 256, 256, 0, stream>>>((float*)d_ws, zc);

    dim3 g1(NB * HH, LL / CHUNK);
    la_phase1<<<g1, 128, 0, stream>>>(K, V, Aws, KS);

    dim3 g2(NB * HH, LL / 64);
    la_phase2<<<g2, 128, 0, stream>>>(Q, Aws, KS, out);
}